// FCCaps_39565238731182
// MI455X (gfx1250) — compile-verified
//
#include <hip/hip_runtime.h>
#include <hip/hip_bf16.h>
#include <math.h>

typedef __attribute__((ext_vector_type(2))) float v2f;
typedef __attribute__((ext_vector_type(8))) float v8f;

#define BATCH 32
#define N_INCAP 1024
#define C_INCH 128
#define NCLS 20
#define DOUT 64
#define CD 1280  // NCLS*DOUT

// ---------------------------------------------------------------------------
// Stage 1: u_hat[b,n,c,d] = sum_i x[b,n,i] * w1[n,c,d,i]
// Per n: [32 x 128] @ [128 x 1280] using V_WMMA_F32_16X16X4_F32.
// grid = (cdTile=5, n=1024), block = 256 (8 waves); wave -> 32x32 output patch.
// ---------------------------------------------------------------------------
__global__ __launch_bounds__(256) void gemm_uhat(const float* __restrict__ x,
                                                 const float* __restrict__ w1,
                                                 float* __restrict__ uhat) {
  __shared__ float xs[BATCH][C_INCH + 4];  // row stride 132 floats: conflict-free frag reads
  const int n   = blockIdx.y;
  const int cd0 = blockIdx.x * 256;
  const int tid = threadIdx.x;

  // Cooperative load of A-tile x[:, n, :]  (32 x 128 = 16 KB)
  for (int idx = tid; idx < BATCH * C_INCH; idx += 256) {
    const int b = idx >> 7;
    const int i = idx & 127;
    xs[b][i] = x[((size_t)b * N_INCAP + n) * C_INCH + i];
  }
  __syncthreads();

  const int wave  = tid >> 5;
  const int lane  = tid & 31;
  const int lhalf = lane >> 4;   // 0: K=k0,k0+1   1: K=k0+2,k0+3
  const int l16   = lane & 15;

  // Wave's two B-fragment columns (cd indices)
  const int cdA = cd0 + wave * 32 + l16;
  const int cdB = cdA + 16;
  const float* __restrict__ w1n  = w1 + (size_t)n * CD * C_INCH;
  const float* __restrict__ wrA  = w1n + (size_t)cdA * C_INCH + 2 * lhalf;
  const float* __restrict__ wrB  = w1n + (size_t)cdB * C_INCH + 2 * lhalf;
  const float* __restrict__ xr0  = &xs[l16][2 * lhalf];        // M-tile 0 (b = 0..15)
  const float* __restrict__ xr1  = &xs[16 + l16][2 * lhalf];   // M-tile 1 (b = 16..31)

  v8f acc00 = {}, acc01 = {}, acc10 = {}, acc11 = {};

#pragma unroll 8
  for (int k0 = 0; k0 < C_INCH; k0 += 4) {
    v2f a0, a1, b0, b1;
    a0.x = xr0[k0];  a0.y = xr0[k0 + 1];
    a1.x = xr1[k0];  a1.y = xr1[k0 + 1];
    b0.x = wrA[k0];  b0.y = wrA[k0 + 1];
    b1.x = wrB[k0];  b1.y = wrB[k0 + 1];
    acc00 = __builtin_amdgcn_wmma_f32_16x16x4_f32(false, a0, false, b0, (short)0, acc00, false, false);
    acc01 = __builtin_amdgcn_wmma_f32_16x16x4_f32(false, a0, false, b1, (short)0, acc01, false, false);
    acc10 = __builtin_amdgcn_wmma_f32_16x16x4_f32(false, a1, false, b0, (short)0, acc10, false, false);
    acc11 = __builtin_amdgcn_wmma_f32_16x16x4_f32(false, a1, false, b1, (short)0, acc11, false, false);
  }

  // D layout: lanes 0-15 -> M = j ; lanes 16-31 -> M = j+8 (N = l16)
#pragma unroll
  for (int j = 0; j < 8; ++j) {
    const int b0r = 0 * 16 + lhalf * 8 + j;
    const int b1r = 1 * 16 + lhalf * 8 + j;
    uhat[((size_t)b0r * N_INCAP + n) * CD + cdA] = acc00[j];
    uhat[((size_t)b0r * N_INCAP + n) * CD + cdB] = acc01[j];
    uhat[((size_t)b1r * N_INCAP + n) * CD + cdA] = acc10[j];
    uhat[((size_t)b1r * N_INCAP + n) * CD + cdB] = acc11[j];
  }
}

// ---------------------------------------------------------------------------
// zero init
// ---------------------------------------------------------------------------
__global__ void zero_buf(float* __restrict__ buf, int n) {
  const int i = blockIdx.x * 256 + threadIdx.x;
  if (i < n) buf[i] = 0.0f;
}

// ---------------------------------------------------------------------------
// p[b,n,:] = softmax(b_ij[b,n,:]) over the 20 classes. One (b,n) per thread.
// ---------------------------------------------------------------------------
__global__ __launch_bounds__(256) void softmax_p(const float* __restrict__ bij,
                                                 float* __restrict__ p) {
  const int idx = blockIdx.x * 256 + threadIdx.x;  // b*N_INCAP + n
  if (idx >= BATCH * N_INCAP) return;
  const float* __restrict__ row = bij + (size_t)idx * NCLS;
  float* __restrict__ prow = p + (size_t)idx * NCLS;
  float r[NCLS];
  float m = -1e30f;
#pragma unroll
  for (int c = 0; c < NCLS; ++c) { r[c] = row[c]; m = fmaxf(m, r[c]); }
  float sum = 0.0f;
#pragma unroll
  for (int c = 0; c < NCLS; ++c) { r[c] = __expf(r[c] - m); sum += r[c]; }
  const float inv = 1.0f / sum;
#pragma unroll
  for (int c = 0; c < NCLS; ++c) prow[c] = r[c] * inv;
}

// ---------------------------------------------------------------------------
// s[b,c,d] = (sum_n p[b,n,c]*u_hat[b,n,c,d]) / (sum_n p[b,n,c]);  v = squash(s).
// On last iter also writes poses + activations to d_out.
// grid = (c=20, b=32), block = 256 (64 d-lanes x 4 n-groups)
// ---------------------------------------------------------------------------
__global__ __launch_bounds__(256) void compute_v(const float* __restrict__ uhat,
                                                 const float* __restrict__ p,
                                                 float* __restrict__ v,
                                                 float* __restrict__ out,
                                                 int last) {
  __shared__ float sred[256];
  __shared__ float pred[4];
  __shared__ float ssh[64];
  __shared__ float s2sh;
  const int c = blockIdx.x, b = blockIdx.y;
  const int tid = threadIdx.x;
  const int d = tid & 63, g = tid >> 6;

  const float* __restrict__ ub = uhat + (((size_t)b * N_INCAP) * NCLS + c) * DOUT + d;
  const float* __restrict__ pb = p + ((size_t)b * N_INCAP) * NCLS + c;

  float acc = 0.0f, pacc = 0.0f;
  for (int n = g; n < N_INCAP; n += 4) {
    const float pv = pb[(size_t)n * NCLS];
    acc  += pv * ub[(size_t)n * CD];
    pacc += pv;
  }
  sred[tid] = acc;
  if (d == 0) pred[g] = pacc;
  __syncthreads();

  if (g == 0) {
    const float psum = pred[0] + pred[1] + pred[2] + pred[3];
    const float s_ = (sred[d] + sred[64 + d] + sred[128 + d] + sred[192 + d]) / psum;
    ssh[d] = s_;
  }
  __syncthreads();
  if (tid < 64) sred[tid] = ssh[tid] * ssh[tid];
  __syncthreads();
  if (tid < 32) sred[tid] += sred[tid + 32];
  __syncthreads();
  if (tid < 16) sred[tid] += sred[tid + 16];
  __syncthreads();
  if (tid < 8)  sred[tid] += sred[tid + 8];
  __syncthreads();
  if (tid < 4)  sred[tid] += sred[tid + 4];
  __syncthreads();
  if (tid < 2)  sred[tid] += sred[tid + 2];
  __syncthreads();
  if (tid == 0) s2sh = sred[0] + sred[1];
  __syncthreads();

  if (g == 0) {
    const float s2 = s2sh;
    const float scale = sqrtf(s2) / (0.5f + s2);
    const float vv = scale * ssh[d];
    v[((size_t)b * NCLS + c) * DOUT + d] = vv;
    if (last) {
      out[((size_t)b * NCLS + c) * DOUT + d] = vv;                 // poses [B,C,D,1]
      if (d == 0) out[BATCH * NCLS * DOUT + b * NCLS + c] = s2 / (0.5f + s2);  // |v|
    }
  }
}

// ---------------------------------------------------------------------------
// b_ij[b,n,c] += 1 - sum_d (squash(u_hat)[b,n,c,d] - v[b,c,d])^2
// grid = (n=1024, b=32), block = 64 (d lanes)
// ---------------------------------------------------------------------------
__global__ __launch_bounds__(64) void update_b(const float* __restrict__ uhat,
                                               const float* __restrict__ v,
                                               float* __restrict__ bij) {
  __shared__ float red[64];
  const int n = blockIdx.x, b = blockIdx.y;
  const int d = threadIdx.x;
  const float* __restrict__ ub = uhat + (((size_t)b * N_INCAP + n) * NCLS) * DOUT;
  const float* __restrict__ vb = v + ((size_t)b * NCLS) * DOUT;
  float* __restrict__ bb = bij + ((size_t)b * N_INCAP + n) * NCLS;

  for (int c = 0; c < NCLS; ++c) {
    const float u = ub[c * DOUT + d];
    red[d] = u * u;
    __syncthreads();
    for (int s = 32; s > 0; s >>= 1) {
      if (d < s) red[d] += red[d + s];
      __syncthreads();
    }
    const float s2 = red[0];
    __syncthreads();
    const float scale = sqrtf(s2) / (0.5f + s2);
    const float diff = scale * u - vb[c * DOUT + d];
    red[d] = diff * diff;
    __syncthreads();
    for (int s = 32; s > 0; s >>= 1) {
      if (d < s) red[d] += red[d + s];
      __syncthreads();
    }
    if (d == 0) bb[c] += 1.0f - red[0];
    __syncthreads();
  }
}

// ---------------------------------------------------------------------------
extern "C" void kernel_launch(void* const* d_in, const int* in_sizes, int n_in,
                              void* d_out, int out_size, void* d_ws, size_t ws_size,
                              hipStream_t stream) {
  const float* x  = (const float*)d_in[0];   // [32,1024,128]
  const float* w1 = (const float*)d_in[1];   // [1024,20,64,128]
  float* out = (float*)d_out;                // 40960 poses + 640 activations

  char* ws = (char*)d_ws;
  const size_t UHAT_BYTES = (size_t)BATCH * N_INCAP * CD * 4;     // 167,772,160
  const size_t BIJ_BYTES  = (size_t)BATCH * N_INCAP * NCLS * 4;   //   2,621,440
  float* uhat = (float*)ws;
  float* bij  = (float*)(ws + UHAT_BYTES);
  float* p    = (float*)(ws + UHAT_BYTES + BIJ_BYTES);
  float* v    = (float*)(ws + UHAT_BYTES + 2 * BIJ_BYTES);

  // Stage 1: WMMA GEMM -> u_hat (HBM-bound on streaming w1 once)
  gemm_uhat<<<dim3(5, N_INCAP), 256, 0, stream>>>(x, w1, uhat);

  // b_ij = 0
  const int bijN = BATCH * N_INCAP * NCLS;
  zero_buf<<<(bijN + 255) / 256, 256, 0, stream>>>(bij, bijN);

  // Stage 2: 3 routing iterations (u_hat is L2-resident: 168 MB < 192 MB L2)
  for (int it = 0; it < 3; ++it) {
    softmax_p<<<(BATCH * N_INCAP) / 256, 256, 0, stream>>>(bij, p);
    compute_v<<<dim3(NCLS, BATCH), 256, 0, stream>>>(uhat, p, v, out, it == 2 ? 1 : 0);
    if (it < 2) {
      update_b<<<dim3(N_INCAP, BATCH), 64, 0, stream>>>(uhat, v, bij);
    }
  }
}